// SeparateSingleLSTM_38173669327132
// MI455X (gfx1250) — compile-verified
//
#include <hip/hip_runtime.h>
#include <math.h>

typedef float v2f __attribute__((ext_vector_type(2)));
typedef float v8f __attribute__((ext_vector_type(8)));

#define BATCH 16
#define SEQ   128
#define NROW  (BATCH * SEQ)      // 2048 rows
#define CH    3
#define IMH   224
#define IMW   224
#define HH    112                // half height/width (quadrant size)
#define POOLD 15
#define EMBD  8

__device__ __forceinline__ int imin(int a, int b) { return a < b ? a : b; }
__device__ __forceinline__ float sigmoidf(float x) { return 1.0f / (1.0f + expf(-x)); }

// ---------------------------------------------------------------------------
// Kernel 1: SPP adaptive max pool (levels [2,1]).
// One block per (image n, channel c). Each quadrant (112x112) reduced with
// float4 loads (rows are 448B-aligned), then LDS tree reduction.
// Writes pooled row layout: [c*4 + h2*2 + w2] for level-2, [12+c] for level-1.
// ---------------------------------------------------------------------------
__global__ __launch_bounds__(256) void spp_kernel(const float* __restrict__ frames,
                                                  float* __restrict__ pooled) {
  const int n = blockIdx.x / CH;
  const int c = blockIdx.x % CH;
  const float* img = frames + ((size_t)n * CH + c) * (size_t)(IMH * IMW);
  const int tid = threadIdx.x;

  __shared__ float red[4][256];

  float qmax[4];
#pragma unroll
  for (int q = 0; q < 4; ++q) {
    const int qr = q >> 1, qc = q & 1;
    const float* qbase = img + qr * HH * IMW + qc * HH;
    float m = -3.402823466e38f;
    // quadrant: 112 rows x 28 float4
    for (int i = tid; i < HH * (HH / 4); i += 256) {
      const int r  = i / (HH / 4);
      const int c4 = i % (HH / 4);
      const float4 v = ((const float4*)(qbase + r * IMW))[c4];
      m = fmaxf(m, fmaxf(fmaxf(v.x, v.y), fmaxf(v.z, v.w)));
    }
    qmax[q] = m;
  }
#pragma unroll
  for (int q = 0; q < 4; ++q) red[q][tid] = qmax[q];
  __syncthreads();
  for (int s = 128; s > 0; s >>= 1) {
    if (tid < s) {
#pragma unroll
      for (int q = 0; q < 4; ++q)
        red[q][tid] = fmaxf(red[q][tid], red[q][tid + s]);
    }
    __syncthreads();
  }
  if (tid == 0) {
    float* row = pooled + (size_t)n * POOLD;
    float all = -3.402823466e38f;
#pragma unroll
    for (int q = 0; q < 4; ++q) {
      row[c * 4 + q] = red[q][0];
      all = fmaxf(all, red[q][0]);
    }
    row[12 + c] = all;
  }
}

// ---------------------------------------------------------------------------
// WMMA fragment loaders for V_WMMA_F32_16X16X4_F32.
// A (16x4 f32): lane m = lane&15; VGPR0/1 hold K = kc*4 + (lane>>4)*2 + {0,1}.
// B (4x16 f32): lane n = lane&15; same K striping. B[k][n] = W[n][k] (W.T).
// Out-of-range (K==15 pad, N >= valid cols) handled with clamp + 0/1 multiply
// so control flow stays uniform (EXEC must be all-1s at the WMMA).
// ---------------------------------------------------------------------------
__device__ __forceinline__ v2f load_a_frag(const float* __restrict__ arow, int kc, int g) {
  const int k0 = kc * 4 + g * 2;
  v2f a;
  a.x = arow[imin(k0, 14)]     * (k0     < POOLD ? 1.0f : 0.0f);
  a.y = arow[imin(k0 + 1, 14)] * (k0 + 1 < POOLD ? 1.0f : 0.0f);
  return a;
}

__device__ __forceinline__ v2f load_b_frag(const float* __restrict__ W, int nvalid,
                                           int n, int kc, int g) {
  const int k0 = kc * 4 + g * 2;
  const float mn = (n < nvalid) ? 1.0f : 0.0f;
  const int nn = imin(n, nvalid - 1);
  v2f b;
  b.x = W[nn * POOLD + imin(k0, 14)]     * mn * (k0     < POOLD ? 1.0f : 0.0f);
  b.y = W[nn * POOLD + imin(k0 + 1, 14)] * mn * (k0 + 1 < POOLD ? 1.0f : 0.0f);
  return b;
}

// ---------------------------------------------------------------------------
// Kernel 2a: forecasted = pooled @ Wf.T + bf  (written to fpooled output)
//            emb        = pooled @ We.T + be  (written to workspace)
// One wave per 16-row tile; 4 chained f32 WMMAs per product over K=16.
// ---------------------------------------------------------------------------
__global__ __launch_bounds__(32) void gemm_fe_kernel(const float* __restrict__ pooled,
                                                     const float* __restrict__ Wf,
                                                     const float* __restrict__ bf,
                                                     const float* __restrict__ We,
                                                     const float* __restrict__ be,
                                                     float* __restrict__ fpooled,
                                                     float* __restrict__ emb) {
  const int lane = threadIdx.x & 31;
  const int g = lane >> 4;
  const int n = lane & 15;             // N index for B/C/D; also M index for A loads
  const int m0 = blockIdx.x * 16;
  const float* arow = pooled + (size_t)(m0 + n) * POOLD;

  v2f a[4], bwf[4], bwe[4];
#pragma unroll
  for (int kc = 0; kc < 4; ++kc) {
    a[kc]   = load_a_frag(arow, kc, g);
    bwf[kc] = load_b_frag(Wf, 15, n, kc, g);
    bwe[kc] = load_b_frag(We, 8, n, kc, g);
  }
  const float bfn = bf[imin(n, 14)] * (n < 15 ? 1.0f : 0.0f);
  const float ben = be[imin(n, 7)]  * (n < 8  ? 1.0f : 0.0f);
  v8f cf = {bfn, bfn, bfn, bfn, bfn, bfn, bfn, bfn};
  v8f ce = {ben, ben, ben, ben, ben, ben, ben, ben};

#pragma unroll
  for (int kc = 0; kc < 4; ++kc) {
    cf = __builtin_amdgcn_wmma_f32_16x16x4_f32(false, a[kc], false, bwf[kc],
                                               (short)0, cf, false, false);
    ce = __builtin_amdgcn_wmma_f32_16x16x4_f32(false, a[kc], false, bwe[kc],
                                               (short)0, ce, false, false);
  }

#pragma unroll
  for (int r = 0; r < 8; ++r) {
    const int row = m0 + r + 8 * g;    // D layout: VGPR r -> M=r (lanes 0-15), M=r+8 (16-31)
    if (n < 15) fpooled[(size_t)row * POOLD + n] = cf[r];
    if (n < 8)  emb[(size_t)row * EMBD + n]      = ce[r];
  }
}

// ---------------------------------------------------------------------------
// Kernel 2b: femb = forecasted @ We.T + be  (reads fpooled written by 2a)
// ---------------------------------------------------------------------------
__global__ __launch_bounds__(32) void gemm_femb_kernel(const float* __restrict__ fpooled,
                                                       const float* __restrict__ We,
                                                       const float* __restrict__ be,
                                                       float* __restrict__ femb) {
  const int lane = threadIdx.x & 31;
  const int g = lane >> 4;
  const int n = lane & 15;
  const int m0 = blockIdx.x * 16;
  const float* arow = fpooled + (size_t)(m0 + n) * POOLD;

  v2f a[4], bwe[4];
#pragma unroll
  for (int kc = 0; kc < 4; ++kc) {
    a[kc]   = load_a_frag(arow, kc, g);
    bwe[kc] = load_b_frag(We, 8, n, kc, g);
  }
  const float ben = be[imin(n, 7)] * (n < 8 ? 1.0f : 0.0f);
  v8f ce = {ben, ben, ben, ben, ben, ben, ben, ben};
#pragma unroll
  for (int kc = 0; kc < 4; ++kc) {
    ce = __builtin_amdgcn_wmma_f32_16x16x4_f32(false, a[kc], false, bwe[kc],
                                               (short)0, ce, false, false);
  }
#pragma unroll
  for (int r = 0; r < 8; ++r) {
    const int row = m0 + r + 8 * g;
    if (n < 8) femb[(size_t)row * EMBD + n] = ce[r];
  }
}

// ---------------------------------------------------------------------------
// Kernel 3: double-cell LSTM scan (hidden=4), sequential over S=128.
// 16 active lanes, one batch element each; weights staged in LDS.
// Gate order (torch): i, f, g, o.
// ---------------------------------------------------------------------------
__device__ __forceinline__ void lstm_cell(const float* __restrict__ x,
                                          const float* h, const float* c,
                                          const float* __restrict__ sWih,
                                          const float* __restrict__ sWhh,
                                          const float* __restrict__ sb,
                                          float* h2, float* c2) {
  float gates[16];
#pragma unroll
  for (int gi = 0; gi < 16; ++gi) {
    float acc = sb[gi];
#pragma unroll
    for (int k = 0; k < 8; ++k) acc += x[k] * sWih[gi * 8 + k];
#pragma unroll
    for (int k = 0; k < 4; ++k) acc += h[k] * sWhh[gi * 4 + k];
    gates[gi] = acc;
  }
#pragma unroll
  for (int j = 0; j < 4; ++j) {
    const float ig = sigmoidf(gates[j]);
    const float fg = sigmoidf(gates[4 + j]);
    const float gg = tanhf(gates[8 + j]);
    const float og = sigmoidf(gates[12 + j]);
    c2[j] = fg * c[j] + ig * gg;
    h2[j] = og * tanhf(c2[j]);
  }
}

__global__ __launch_bounds__(32) void lstm_kernel(const float* __restrict__ emb,
                                                  const float* __restrict__ femb,
                                                  const float* __restrict__ Wih,
                                                  const float* __restrict__ Whh,
                                                  const float* __restrict__ bih,
                                                  const float* __restrict__ bhh,
                                                  const float* __restrict__ Wp,
                                                  const float* __restrict__ bp,
                                                  float* __restrict__ prog,
                                                  float* __restrict__ fprog) {
  __shared__ float sWih[16 * 8];
  __shared__ float sWhh[16 * 4];
  __shared__ float sb[16];
  __shared__ float sWp[4];
  __shared__ float sbp;
  const int tid = threadIdx.x;
  for (int i = tid; i < 128; i += 32) sWih[i] = Wih[i];
  for (int i = tid; i < 64;  i += 32) sWhh[i] = Whh[i];
  if (tid < 16) sb[tid] = bih[tid] + bhh[tid];
  if (tid < 4)  sWp[tid] = Wp[tid];
  if (tid == 0) sbp = bp[0];
  __syncthreads();
  if (tid >= BATCH) return;

  const int b = tid;
  float h[4] = {0.f, 0.f, 0.f, 0.f};
  float c[4] = {0.f, 0.f, 0.f, 0.f};
  for (int s = 0; s < SEQ; ++s) {
    const int row = b * SEQ + s;
    float h1[4], c1[4], h2[4], c2[4];
    lstm_cell(&emb[(size_t)row * EMBD], h, c, sWih, sWhh, sb, h1, c1);
    lstm_cell(&femb[(size_t)row * EMBD], h1, c1, sWih, sWhh, sb, h2, c2);

    const float p  = h1[0] * sWp[0] + h1[1] * sWp[1] + h1[2] * sWp[2] + h1[3] * sWp[3] + sbp;
    const float fp = h2[0] * sWp[0] + h2[1] * sWp[1] + h2[2] * sWp[2] + h2[3] * sWp[3] + sbp;
    prog[row]  = sigmoidf(p);
    fprog[row] = sigmoidf(fp);

#pragma unroll
    for (int j = 0; j < 4; ++j) { h[j] = h1[j]; c[j] = c1[j]; }
  }
}

// ---------------------------------------------------------------------------
// d_out layout (flat, fp32): progress (2048) | forecasted_progress (2048) |
//                            pooled (2048*15) | forecasted_pooled (2048*15)
// d_ws: emb (2048*8 f32) | femb (2048*8 f32)  = 128 KB
// ---------------------------------------------------------------------------
extern "C" void kernel_launch(void* const* d_in, const int* in_sizes, int n_in,
                              void* d_out, int out_size, void* d_ws, size_t ws_size,
                              hipStream_t stream) {
  (void)in_sizes; (void)n_in; (void)out_size; (void)ws_size;
  const float* frames = (const float*)d_in[0];
  const float* Wf  = (const float*)d_in[1];
  const float* bf  = (const float*)d_in[2];
  const float* We  = (const float*)d_in[3];
  const float* be  = (const float*)d_in[4];
  const float* Wih = (const float*)d_in[5];
  const float* Whh = (const float*)d_in[6];
  const float* bih = (const float*)d_in[7];
  const float* bhh = (const float*)d_in[8];
  const float* Wp  = (const float*)d_in[9];
  const float* bp  = (const float*)d_in[10];

  float* out     = (float*)d_out;
  float* prog    = out;                       // (16,128)
  float* fprog   = out + NROW;                // (16,128)
  float* pooled  = out + 2 * NROW;            // (2048,15)
  float* fpooled = pooled + NROW * POOLD;     // (2048,15)

  float* emb  = (float*)d_ws;                 // (2048,8)
  float* femb = emb + NROW * EMBD;            // (2048,8)

  spp_kernel<<<dim3(NROW * CH), dim3(256), 0, stream>>>(frames, pooled);
  gemm_fe_kernel<<<dim3(NROW / 16), dim3(32), 0, stream>>>(pooled, Wf, bf, We, be,
                                                           fpooled, emb);
  gemm_femb_kernel<<<dim3(NROW / 16), dim3(32), 0, stream>>>(fpooled, We, be, femb);
  lstm_kernel<<<dim3(1), dim3(32), 0, stream>>>(emb, femb, Wih, Whh, bih, bhh,
                                                Wp, bp, prog, fprog);
}